// HeteroSTHN_Interface_13769665151132
// MI455X (gfx1250) — compile-verified
//
#include <hip/hip_runtime.h>

// ---------------- problem constants (from reference setup_inputs) ------------
constexpr int E   = 65536;   // edges
constexpr int D   = 200;     // feature dim (K)  = 50 k-steps of 4
constexpr int H   = 100;     // hidden per type
constexpr int T   = 8;       // edge types
constexpr int HP  = 112;     // H padded to 7*16
constexpr int NT  = 7;       // n-tiles of 16 per type
constexpr int KS  = 50;      // D / 4 k-steps
constexpr float NEG_INF  = -1e30f;
constexpr float VALID_TH = -1e29f;

// Each (type, sel) weight blob: KS*NT*32 fragments * 8B = 89,600 B.
// Padded to 90,112 B = 5632 x b128 so every thread moves exactly 22 x b128.
constexpr int BLOB_B128   = 5632;                 // b128 transfers per blob
constexpr int BLOB_BYTES  = BLOB_B128 * 16;       // 90,112
constexpr int BLOB_FLOATS = BLOB_BYTES / 4;       // 22,528
constexpr int XFER_PER_THREAD = BLOB_B128 / 256;  // 22 (uniform)

// packed-weight workspace layout (floats)
constexpr int WOFF_BIAS = 2 * T * BLOB_FLOATS;    // 360,448
constexpr int WOFF_OUTW = WOFF_BIAS + T * HP;
constexpr int WOFF_OUTB = WOFF_OUTW + T * HP;

typedef __attribute__((ext_vector_type(2))) float v2f;
typedef __attribute__((ext_vector_type(8))) float v8f;

#if __has_builtin(__builtin_amdgcn_global_load_async_to_lds_b128)
#define HAVE_ASYNC_LDS 1
typedef int v4i_t __attribute__((ext_vector_type(4)));
typedef __attribute__((address_space(1))) v4i_t gv4i;   // global int4
typedef __attribute__((address_space(3))) v4i_t lv4i;   // LDS int4
#else
#define HAVE_ASYNC_LDS 0
#endif

template<int N>
__device__ inline void wait_asynccnt() {
#if HAVE_ASYNC_LDS
#if __has_builtin(__builtin_amdgcn_s_wait_asynccnt)
  __builtin_amdgcn_s_wait_asynccnt(N);
#else
  asm volatile("s_wait_asynccnt %0" :: "i"(N));
#endif
#endif
}

// Stage one 90,112 B weight blob global->LDS. Async path uses the CDNA5
// GLOBAL_LOAD_ASYNC_TO_LDS_B128 (ASYNCcnt) so the copy runs behind WMMA math.
__device__ inline void stage_blob(const float* __restrict__ gsrc, void* ldst, int tid) {
#if HAVE_ASYNC_LDS
  gv4i* g = (gv4i*)gsrc;
  lv4i* s = (lv4i*)ldst;
  #pragma unroll
  for (int u = 0; u < XFER_PER_THREAD; ++u) {
    int i = tid + u * 256;
    __builtin_amdgcn_global_load_async_to_lds_b128(g + i, s + i, 0, 0);
  }
#else
  const float4* g = (const float4*)gsrc;
  float4* s = (float4*)ldst;
  #pragma unroll
  for (int u = 0; u < XFER_PER_THREAD; ++u) {
    int i = tid + u * 256;
    s[i] = g[i];
  }
#endif
}

// -------------------------------------------------------------------------
// Kernel 0: pack weights into WMMA B-fragment layout.
//   v0: lane<16 -> B[K=0,N=lane], lane>=16 -> B[K=1,N=lane-16]
//   v1:          B[K=2,N],                     B[K=3,N]
// Blob (t,sel) at ws[(t*2+sel)*BLOB_FLOATS], fragment (k,j,lane) at
// ((k*NT+j)*32+lane)*2. Also packs fused bias, padded out_W, out_b.
// -------------------------------------------------------------------------
__global__ void pack_weights_kernel(const float* __restrict__ srcW,
                                    const float* __restrict__ srcB,
                                    const float* __restrict__ dstW,
                                    const float* __restrict__ dstB,
                                    const float* __restrict__ outW,
                                    const float* __restrict__ outB,
                                    float* __restrict__ ws) {
  const int tid = blockIdx.x * blockDim.x + threadIdx.x;
  const int NW = 2 * T * KS * NT * 32;   // 179200 fragment slots
  if (tid < NW) {
    int lane = tid & 31;
    int rest = tid >> 5;
    int j = rest % NT; rest /= NT;
    int k = rest % KS; rest /= KS;
    int sel = rest & 1;
    int t   = rest >> 1;
    int hf  = lane >> 4;
    int col = lane & 15;
    int n   = j * 16 + col;
    int d0  = 4 * k + hf;       // v0 row
    int d1  = d0 + 2;           // v1 row
    const float* W = sel ? dstW : srcW;   // (T, D, H) row-major
    float v0 = 0.f, v1 = 0.f;
    if (n < H) {
      v0 = W[(t * D + d0) * H + n];
      v1 = W[(t * D + d1) * H + n];
    }
    float* dst = ws + (size_t)(t * 2 + sel) * BLOB_FLOATS + ((k * NT + j) * 32 + lane) * 2;
    dst[0] = v0;
    dst[1] = v1;
  } else {
    int u = tid - NW;
    if (u < T * HP) {                     // combined bias, padded
      int t = u / HP, hh = u % HP;
      ws[WOFF_BIAS + u] = (hh < H) ? (srcB[t * H + hh] + dstB[t * H + hh]) : 0.f;
    } else if (u < 2 * T * HP) {          // out_W padded with zeros (kills pad cols)
      int u2 = u - T * HP;
      int t = u2 / HP, hh = u2 % HP;
      ws[WOFF_OUTW + u2] = (hh < H) ? outW[t * H + hh] : 0.f;
    } else if (u < 2 * T * HP + T) {
      int t = u - 2 * T * HP;
      ws[WOFF_OUTB + t] = outB[t];
    }
  }
}

// -------------------------------------------------------------------------
// Kernel 1: fused edge scorer. Block = 8 waves, one 16-edge tile per wave.
// LDS double-buffered at half-blob granularity:
//   region A (src weights) free during dst phase -> async-prefetch src(t+1)
//   region B (dst weights) free during src phase -> async-prefetch dst(t+1)
// Residency via in-order ASYNCcnt: s_wait_asynccnt 22 + barrier.
// -------------------------------------------------------------------------
__global__ __launch_bounds__(256)
void hetero_pred_kernel(const float* __restrict__ h,
                        const int*   __restrict__ pe,
                        const float* __restrict__ ws,
                        float*       __restrict__ out) {
  __shared__ __attribute__((aligned(16))) char wlds[2 * BLOB_BYTES];  // 180,224 B

  const int tid  = threadIdx.x;
  const int lane = tid & 31;
  const int wave = tid >> 5;
  const int col  = lane & 15;
  const int hf   = lane >> 4;
  const int e0   = (blockIdx.x * 8 + wave) * 16;

  const v2f* __restrict__ ldsA = (const v2f*)wlds;                 // src weights
  const v2f* __restrict__ ldsB = (const v2f*)(wlds + BLOB_BYTES);  // dst weights

  // A fragment pointers: lane reads h[row, 4k + 2*hf + {0,1}]
  const v2f* __restrict__ aSrc = (const v2f*)(h + (size_t)(e0 + col) * D         + 2 * hf);
  const v2f* __restrict__ aPos = (const v2f*)(h + (size_t)(E + e0 + col) * D     + 2 * hf);
  const v2f* __restrict__ aNeg = (const v2f*)(h + (size_t)(2 * E + e0 + col) * D + 2 * hf);

  float runp[8], runn[8];
  #pragma unroll
  for (int r = 0; r < 8; ++r) { runp[r] = NEG_INF; runn[r] = NEG_INF; }

  // ---- prologue: stage type-0 src + dst blobs ----
  stage_blob(ws + 0 * BLOB_FLOATS, (void*)ldsA, tid);
  stage_blob(ws + 1 * BLOB_FLOATS, (void*)ldsB, tid);
  wait_asynccnt<0>();
  __syncthreads();

  for (int t = 0; t < T; ++t) {
    // ---- src phase: sacc = h_src x srcW (region A) ----
    v8f sacc[NT];
    #pragma unroll
    for (int j = 0; j < NT; ++j) sacc[j] = (v8f){0.f,0.f,0.f,0.f,0.f,0.f,0.f,0.f};

    for (int k = 0; k < KS; ++k) {
      v2f a = aSrc[2 * k];
      const v2f* bp = &ldsA[(k * NT) * 32 + lane];
      #pragma unroll
      for (int j = 0; j < NT; ++j) {
        v2f b = bp[j * 32];
        sacc[j] = __builtin_amdgcn_wmma_f32_16x16x4_f32(
            false, a, false, b, (short)0, sacc[j], false, false);
      }
    }
    __syncthreads();                               // all waves done reading A

    if (t < T - 1)                                 // prefetch src(t+1) under dst phase
      stage_blob(ws + (size_t)((t + 1) * 2 + 0) * BLOB_FLOATS, (void*)ldsA, tid);
    // ensure dst(t) resident: it precedes the 22 just-issued A transfers in order
    if (t < T - 1) wait_asynccnt<XFER_PER_THREAD>(); else wait_asynccnt<0>();
    __syncthreads();

    // ---- dst phase: pos continues on sacc, neg on a copy; B frag shared ----
    v8f nacc[NT];
    #pragma unroll
    for (int j = 0; j < NT; ++j) nacc[j] = sacc[j];

    for (int k = 0; k < KS; ++k) {
      v2f ap = aPos[2 * k];
      v2f an = aNeg[2 * k];
      const v2f* bp = &ldsB[(k * NT) * 32 + lane];
      #pragma unroll
      for (int j = 0; j < NT; ++j) {
        v2f b = bp[j * 32];
        sacc[j] = __builtin_amdgcn_wmma_f32_16x16x4_f32(
            false, ap, false, b, (short)0, sacc[j], false, false);
        nacc[j] = __builtin_amdgcn_wmma_f32_16x16x4_f32(
            false, an, false, b, (short)0, nacc[j], false, false);
      }
    }

    // ---- epilogue for this type ----
    float biasv[NT], oww[NT];
    #pragma unroll
    for (int j = 0; j < NT; ++j) {
      biasv[j] = ws[WOFF_BIAS + t * HP + j * 16 + col];
      oww[j]   = ws[WOFF_OUTW + t * HP + j * 16 + col];
    }
    const float ob = ws[WOFF_OUTB + t];

    #pragma unroll
    for (int r = 0; r < 8; ++r) {
      // C/D layout: VGPR r holds row M=r (lanes 0-15) and M=r+8 (lanes 16-31)
      float sp = 0.f, sn = 0.f;
      #pragma unroll
      for (int j = 0; j < NT; ++j) {
        float hp = fmaxf(sacc[j][r] + biasv[j], 0.f);
        float hn = fmaxf(nacc[j][r] + biasv[j], 0.f);
        sp = fmaf(hp, oww[j], sp);
        sn = fmaf(hn, oww[j], sn);
      }
      #pragma unroll
      for (int m = 1; m <= 8; m <<= 1) {   // reduce within each 16-lane half
        sp += __shfl_xor(sp, m, 32);
        sn += __shfl_xor(sn, m, 32);
      }
      sp += ob; sn += ob;
      const int ei = e0 + 8 * hf + r;
      const bool mp = pe[(size_t)ei * T + t] != 0;
      const bool mn = pe[(size_t)(E + ei) * T + t] != 0;
      runp[r] = fmaxf(runp[r], mp ? sp : NEG_INF);
      runn[r] = fmaxf(runn[r], mn ? sn : NEG_INF);
    }
    __syncthreads();                               // all waves done reading B

    if (t < T - 1) {
      stage_blob(ws + (size_t)((t + 1) * 2 + 1) * BLOB_FLOATS, (void*)ldsB, tid);
      wait_asynccnt<XFER_PER_THREAD>();            // ensure src(t+1) resident
      __syncthreads();
    }
  }

  // ---- final masked-max results; lane 0 / lane 16 write rows r / r+8 ----
  if (col == 0) {
    #pragma unroll
    for (int r = 0; r < 8; ++r) {
      const int ei = e0 + 8 * hf + r;
      out[ei]     = (runp[r] > VALID_TH) ? runp[r] : 0.f;
      out[E + ei] = (runn[r] > VALID_TH) ? runn[r] : 0.f;
    }
  }
}

// -------------------------------------------------------------------------
extern "C" void kernel_launch(void* const* d_in, const int* in_sizes, int n_in,
                              void* d_out, int out_size, void* d_ws, size_t ws_size,
                              hipStream_t stream) {
  const float* h    = (const float*)d_in[0];
  const float* srcW = (const float*)d_in[1];
  const float* srcB = (const float*)d_in[2];
  const float* dstW = (const float*)d_in[3];
  const float* dstB = (const float*)d_in[4];
  const float* outW = (const float*)d_in[5];
  const float* outB = (const float*)d_in[6];
  const int*   pe   = (const int*)d_in[7];
  float* ws  = (float*)d_ws;
  float* out = (float*)d_out;

  const int packThreads = 2 * T * KS * NT * 32 + 2 * T * HP + T;
  const int packBlocks  = (packThreads + 255) / 256;
  pack_weights_kernel<<<packBlocks, 256, 0, stream>>>(srcW, srcB, dstW, dstB,
                                                      outW, outB, ws);
  // 4096 edge tiles of 16, 8 waves (tiles) per block -> 512 blocks
  hetero_pred_kernel<<<dim3(E / (16 * 8)), 256, 0, stream>>>(h, pe, ws, out);
}